// GaussianMaskRenderer_23673859735673
// MI455X (gfx1250) — compile-verified
//
#include <hip/hip_runtime.h>
#include <math.h>

#define NG   6144         // gaussians per (b,t) render
#define NR   6            // (b*t) * views = 2*3 renders
#define HH   224
#define WW   448
#define SPIX (HH*WW)
#define RADI 2
#define CH   256          // gaussians staged per LDS chunk
#define NCH  (NG/CH)      // 24 chunks

typedef __attribute__((ext_vector_type(2))) float v2f;
typedef __attribute__((ext_vector_type(8))) float v8f;

// ---------------- workspace layout (float units) ----------------
#define OFF_AOS   0                        // NR*NG*12  (16B aligned, first)
#define OFF_MATS  (OFF_AOS + NR*NG*12)     // NR*16
#define OFF_U     (OFF_MATS + NR*16)       // NR*NG
#define OFF_V     (OFF_U + NR*NG)          // NR*NG
#define OFF_SIG   (OFF_V + NR*NG)          // NR*NG
#define OFF_WSIG  (OFF_SIG + NR*NG)        // NR*NG
#define OFF_WCNT  (OFF_WSIG + NR*NG)       // NR*NG
#define OFF_KEYS  (OFF_WCNT + NR*NG)       // NR*NG
#define OFF_SIDX  (OFF_KEYS + NR*NG)       // NR*NG ints
#define OFF_TOUCH (OFF_SIDX + NR*NG)       // 1 unsigned

__device__ __forceinline__ float clamp01(float x){ return fminf(fmaxf(x,0.0f),1.0f); }

// ---------------- kernel: zero accumulators ----------------
__global__ void k_zero(unsigned* __restrict__ touch){
  if (threadIdx.x == 0) *touch = 0u;
}

// ---------------- kernel: build combined matrices M = inv(c2w) * first_pose ----
__global__ void k_mats(const float* __restrict__ c2w, const float* __restrict__ fp,
                       float* __restrict__ mats){
  int r = threadIdx.x;
  if (r >= NR) return;
  const float* m = c2w + r*16;
  float inv[16];
  inv[0]  =  m[5]*m[10]*m[15] - m[5]*m[11]*m[14] - m[9]*m[6]*m[15] + m[9]*m[7]*m[14] + m[13]*m[6]*m[11] - m[13]*m[7]*m[10];
  inv[4]  = -m[4]*m[10]*m[15] + m[4]*m[11]*m[14] + m[8]*m[6]*m[15] - m[8]*m[7]*m[14] - m[12]*m[6]*m[11] + m[12]*m[7]*m[10];
  inv[8]  =  m[4]*m[9]*m[15]  - m[4]*m[11]*m[13] - m[8]*m[5]*m[15] + m[8]*m[7]*m[13] + m[12]*m[5]*m[11] - m[12]*m[7]*m[9];
  inv[12] = -m[4]*m[9]*m[14]  + m[4]*m[10]*m[13] + m[8]*m[5]*m[14] - m[8]*m[6]*m[13] - m[12]*m[5]*m[10] + m[12]*m[6]*m[9];
  inv[1]  = -m[1]*m[10]*m[15] + m[1]*m[11]*m[14] + m[9]*m[2]*m[15] - m[9]*m[3]*m[14] - m[13]*m[2]*m[11] + m[13]*m[3]*m[10];
  inv[5]  =  m[0]*m[10]*m[15] - m[0]*m[11]*m[14] - m[8]*m[2]*m[15] + m[8]*m[3]*m[14] + m[12]*m[2]*m[11] - m[12]*m[3]*m[10];
  inv[9]  = -m[0]*m[9]*m[15]  + m[0]*m[11]*m[13] + m[8]*m[1]*m[15] - m[8]*m[3]*m[13] - m[12]*m[1]*m[11] + m[12]*m[3]*m[9];
  inv[13] =  m[0]*m[9]*m[14]  - m[0]*m[10]*m[13] - m[8]*m[1]*m[14] + m[8]*m[2]*m[13] + m[12]*m[1]*m[10] - m[12]*m[2]*m[9];
  inv[2]  =  m[1]*m[6]*m[15]  - m[1]*m[7]*m[14]  - m[5]*m[2]*m[15] + m[5]*m[3]*m[14] + m[13]*m[2]*m[7]  - m[13]*m[3]*m[6];
  inv[6]  = -m[0]*m[6]*m[15]  + m[0]*m[7]*m[14]  + m[4]*m[2]*m[15] - m[4]*m[3]*m[14] - m[12]*m[2]*m[7]  + m[12]*m[3]*m[6];
  inv[10] =  m[0]*m[5]*m[15]  - m[0]*m[7]*m[13]  - m[4]*m[1]*m[15] + m[4]*m[3]*m[13] + m[12]*m[1]*m[7]  - m[12]*m[3]*m[5];
  inv[14] = -m[0]*m[5]*m[14]  + m[0]*m[6]*m[13]  + m[4]*m[1]*m[14] - m[4]*m[2]*m[13] - m[12]*m[1]*m[6]  + m[12]*m[2]*m[5];
  inv[3]  = -m[1]*m[6]*m[11]  + m[1]*m[7]*m[10]  + m[5]*m[2]*m[11] - m[5]*m[3]*m[10] - m[9]*m[2]*m[7]   + m[9]*m[3]*m[6];
  inv[7]  =  m[0]*m[6]*m[11]  - m[0]*m[7]*m[10]  - m[4]*m[2]*m[11] + m[4]*m[3]*m[10] + m[8]*m[2]*m[7]   - m[8]*m[3]*m[6];
  inv[11] = -m[0]*m[5]*m[11]  + m[0]*m[7]*m[9]   + m[4]*m[1]*m[11] - m[4]*m[3]*m[9]  - m[8]*m[1]*m[7]   + m[8]*m[3]*m[5];
  inv[15] =  m[0]*m[5]*m[10]  - m[0]*m[6]*m[9]   - m[4]*m[1]*m[10] + m[4]*m[2]*m[9]  + m[8]*m[1]*m[6]   - m[8]*m[2]*m[5];
  float det = m[0]*inv[0] + m[1]*inv[4] + m[2]*inv[8] + m[3]*inv[12];
  float id  = (det != 0.0f) ? 1.0f/det : 0.0f;
  for (int i=0;i<4;i++)
    for (int j=0;j<4;j++){
      float s = 0.0f;
      for (int k=0;k<4;k++) s += inv[i*4+k]*id * fp[k*4+j];
      mats[r*16 + i*4 + j] = s;
    }
}

// ---------------- kernel: WMMA point transform + projection ----------------
// 144 blocks x 256 thr; block = (render r = bx/24, chunk = bx%24);
// each wave transforms 32 points with TWO 16x16x4 f32 WMMAs, then every lane
// projects one point.
__global__ __launch_bounds__(256) void k_project(
    const float* __restrict__ centers, const float* __restrict__ scale,
    const float* __restrict__ opac,    const float* __restrict__ intr,
    const float* __restrict__ mats,
    float* __restrict__ U, float* __restrict__ Vv, float* __restrict__ SIG,
    float* __restrict__ WSIG, float* __restrict__ WCNT, float* __restrict__ KEYS)
{
  int r     = blockIdx.x / 24;
  int chunk = blockIdx.x % 24;
  int tid   = threadIdx.x;
  int wave  = tid >> 5, lane = tid & 31;
  int m     = lane & 15, half = lane >> 4;
  int base  = chunk*256 + wave*32;
  int bt    = r/3, view = r%3;
  const float* Mr = mats + r*16;

  // A operands for two 16-point groups (branch-free: EXEC stays full for WMMA)
  const float* C1 = centers + (size_t)(bt*NG + base + m)*3;
  const float* C2 = C1 + 16*3;
  float x1=C1[0], y1=C1[1], z1=C1[2];
  float x2=C2[0], y2=C2[1], z2=C2[2];

  // B operand: B[k][n] = M[n][k] for n<4, else 0.  Clamped-index load + select
  // (no divergent control flow before the WMMAs).
  int   nc = (m < 4) ? m : 0;
  float b0 = half ? Mr[nc*4+2] : Mr[nc*4+0];
  float b1 = half ? Mr[nc*4+3] : Mr[nc*4+1];
  b0 = (m < 4) ? b0 : 0.0f;
  b1 = (m < 4) ? b1 : 0.0f;
  v2f b; b.x = b0; b.y = b1;

  v2f a1; a1.x = half ? z1 : x1;  a1.y = half ? 1.0f : y1;
  v2f a2; a2.x = half ? z2 : x2;  a2.y = half ? 1.0f : y2;
  v8f c = {0.f,0.f,0.f,0.f,0.f,0.f,0.f,0.f};

  __shared__ float sh[8*512];   // per wave: two 16x16 D tiles (512 floats)
  v8f d1 = __builtin_amdgcn_wmma_f32_16x16x4_f32(false, a1, false, b, (short)0, c, false, false);
  v8f d2 = __builtin_amdgcn_wmma_f32_16x16x4_f32(false, a2, false, b, (short)0, c, false, false);
  #pragma unroll
  for (int i=0;i<8;i++){
    sh[wave*512 +       i*32 + lane] = d1[i];
    sh[wave*512 + 256 + i*32 + lane] = d2[i];
  }
  __syncthreads();

  // lane L owns point (base + L): group = L>>4, row mm = L&15, col = coord
  {
    int grp = lane >> 4, mm = lane & 15;
    int bL  = wave*512 + grp*256 + ((mm < 8) ? mm*32 : (mm-8)*32 + 16);
    float camx = sh[bL+0], camy = sh[bL+1], camz = sh[bL+2];

    int i  = base + lane;
    int gp = bt*NG + i;
    float fx = intr[view*4+0], fy = intr[view*4+1];
    float cx = intr[view*4+2], cy = intr[view*4+3];
    bool fin = isfinite(camx) && isfinite(camy) && isfinite(camz);
    bool gv  = (camz > 0.001f) && fin;
    float zs = gv ? camz : 1.0f;
    float u  = gv ? (camx*fx/zs + cx) : -1.0e6f;
    float v  = gv ? (camy*fy/zs + cy) : -1.0e6f;
    float sf = (scale[gp*3+0] + scale[gp*3+1] + scale[gp*3+2]) * (1.0f/3.0f);
    float sg = fminf(fmaxf((fx+fy)*0.5f*fabsf(sf)/fmaxf(zs,0.001f), 0.75f), 10.0f);
    bool inb = (u >= -(float)RADI-1.0f) && (u <= (float)WW+(float)RADI) &&
               (v >= -(float)RADI-1.0f) && (v <= (float)HH+(float)RADI);
    bool valid = gv && inb;
    float opv  = clamp01(opac[gp]);
    bool  vsm  = valid && (opv > 1e-5f);
    U[r*NG+i]   = u;  Vv[r*NG+i] = v;  SIG[r*NG+i] = sg;
    WSIG[r*NG+i] = vsm ? sg  : 0.0f;
    WCNT[r*NG+i] = vsm ? 1.0f : 0.0f;
    KEYS[r*NG+i] = valid ? camz : INFINITY;
  }
}

// ---------------- kernel: per-render bitonic depth sort (8192 in LDS) --------
__global__ __launch_bounds__(1024) void k_sort(const float* __restrict__ KEYS,
                                               int* __restrict__ SIDX){
  int r = blockIdx.x;
  __shared__ float sk[8192];
  __shared__ int   sv[8192];
  for (int i = threadIdx.x; i < 8192; i += 1024){
    sk[i] = (i < NG) ? KEYS[r*NG + i] : INFINITY;
    sv[i] = i;
  }
  __syncthreads();
  for (int size = 2; size <= 8192; size <<= 1){
    for (int stride = size >> 1; stride > 0; stride >>= 1){
      for (int p = threadIdx.x; p < 4096; p += 1024){
        int i = 2*p - (p & (stride-1));
        int j = i + stride;
        bool up = ((i & size) == 0);
        float ki = sk[i], kj = sk[j];
        if ((ki > kj) == up){
          sk[i] = kj; sk[j] = ki;
          int t = sv[i]; sv[i] = sv[j]; sv[j] = t;
        }
      }
      __syncthreads();
    }
  }
  for (int i = threadIdx.x; i < NG; i += 1024) SIDX[r*NG + i] = sv[i];
}

// ---------------- kernel: gather sorted AoS (48B/gaussian, B128-friendly) ----
__global__ __launch_bounds__(256) void k_gather(
    const int* __restrict__ SIDX,
    const float* __restrict__ U, const float* __restrict__ Vv, const float* __restrict__ SIG,
    const float* __restrict__ feat, const float* __restrict__ opac, const float* __restrict__ bgp,
    float* __restrict__ AOS)
{
  int g = blockIdx.x*256 + threadIdx.x;
  if (g >= NR*NG) return;
  int r = g / NG, i = g % NG;
  int j  = SIDX[r*NG + i];
  int bt = r/3;
  int gp = bt*NG + j;
  float u = U[r*NG+j], v = Vv[r*NG+j], sg = SIG[r*NG+j];
  float cr = clamp01(feat[gp*3+0]);
  float cg = clamp01(feat[gp*3+1]);
  float cb = clamp01(feat[gp*3+2]);
  float opv = clamp01(opac[gp]);
  float dyn = clamp01(1.0f - bgp[gp]);
  float rad = fminf(fmaxf(ceilf(sg * 1.5f), 1.0f), (float)RADI);
  float* o = AOS + (size_t)g*12;
  o[0] = u;  o[1] = v;  o[2] = floorf(u);  o[3] = floorf(v);
  o[4] = 1.0f / fmaxf(sg, 0.001f);         o[5] = rad;
  o[6] = cr; o[7] = cg; o[8] = cb;
  o[9]  = opv * (1.0f - dyn);   // sta
  o[10] = opv * dyn;            // dyn
  o[11] = opv;                  // all
}

// ---------------- async LDS staging helper (3x B128 per lane) ----------------
__device__ __forceinline__ void stage_async(const float* src, unsigned lb, int t){
  #pragma unroll
  for (int k = 0; k < 3; ++k){
    int e = t + k*256;  // float4 index within chunk
    unsigned long long ga = (unsigned long long)(const void*)(src + e*4);
    unsigned la = lb + (unsigned)e * 16u;
    asm volatile("global_load_async_to_lds_b128 %0, %1, off"
                 :: "v"(la), "v"(ga) : "memory");
  }
}

// ---------------- kernel: tile renderer (double-buffered async staging) ------
__global__ __launch_bounds__(256) void k_render(const float* __restrict__ AOS,
                                                float* __restrict__ out,
                                                unsigned* __restrict__ touch)
{
  __shared__ float sh[2][CH*12];    // ping-pong chunk buffers (2 x 12 KB)
  __shared__ unsigned scnt;
  int r  = blockIdx.z;
  int px = blockIdx.x*16 + threadIdx.x;
  int py = blockIdx.y*16 + threadIdx.y;
  int t  = threadIdx.y*16 + threadIdx.x;
  if (t == 0) scnt = 0u;
  float pxf = (float)px, pyf = (float)py;
  float aden[3] = {0.f,0.f,0.f};
  float acc[9]  = {0.f,0.f,0.f,0.f,0.f,0.f,0.f,0.f,0.f};
  const float* base = AOS + (size_t)r*NG*12;
  unsigned lb0 = (unsigned)(unsigned long long)(void*)&sh[0][0];
  unsigned lb1 = (unsigned)(unsigned long long)(void*)&sh[1][0];

  stage_async(base, lb0, t);                       // prologue: chunk 0 -> buf0
  for (int c = 0; c < NCH; ++c){
    if (c + 1 < NCH){
      stage_async(base + (size_t)(c+1)*CH*12, (c & 1) ? lb0 : lb1, t);
      asm volatile("s_wait_asynccnt 3" ::: "memory");   // chunk c landed
    } else {
      asm volatile("s_wait_asynccnt 0" ::: "memory");
    }
    __syncthreads();

    const float* gbuf = sh[c & 1];
    for (int j = 0; j < CH; ++j){
      const float* g = gbuf + j*12;
      float rad = g[5];
      float dxi = pxf - g[2], dyi = pyf - g[3];
      if (fabsf(dxi) > rad || fabsf(dyi) > rad) continue;
      float du = (g[0] - pxf) * g[4];
      float dv = (g[1] - pyf) * g[4];
      float la2 = __expf(-0.5f*(du*du + dv*dv));
      #pragma unroll
      for (int b2 = 0; b2 < 3; ++b2){
        float lab     = fminf(la2 * g[9+b2], 0.999f);
        float contrib = lab * (1.0f - aden[b2]);
        aden[b2] = fminf(aden[b2] + contrib, 0.999f);
        acc[b2*3+0] += g[6]*contrib;
        acc[b2*3+1] += g[7]*contrib;
        acc[b2*3+2] += g[8]*contrib;
      }
    }
    __syncthreads();   // all waves done reading buf[c&1] before it is re-staged
  }

  int pix = py*WW + px;
  #pragma unroll
  for (int b2 = 0; b2 < 3; ++b2){
    #pragma unroll
    for (int ch = 0; ch < 3; ++ch)
      out[(size_t)(b2*18 + r*3 + ch)*SPIX + pix] = clamp01(acc[b2*3+ch]);
    out[(size_t)(54 + b2*6 + r)*SPIX + pix] = clamp01(aden[b2]);
  }
  if (aden[2] > 1e-6f) atomicAdd(&scnt, 1u);
  __syncthreads();
  if (t == 0 && scnt) atomicAdd(touch, scnt);
}

// ---------------- kernel: deterministic scalar stats ----------------
__global__ __launch_bounds__(256) void k_stats(const float* __restrict__ WSIG,
                                               const float* __restrict__ WCNT,
                                               const unsigned* __restrict__ touch,
                                               float* __restrict__ out)
{
  __shared__ float red[256];
  int tid = threadIdx.x;
  float smacc = 0.0f;
  for (int r = 0; r < NR; ++r){
    float s = 0.0f, cN = 0.0f;
    for (int i = tid; i < NG; i += 256){ s += WSIG[r*NG+i]; cN += WCNT[r*NG+i]; }
    red[tid] = s;  __syncthreads();
    for (int o = 128; o > 0; o >>= 1){ if (tid < o) red[tid] += red[tid+o]; __syncthreads(); }
    float ssum = red[0];  __syncthreads();
    red[tid] = cN; __syncthreads();
    for (int o = 128; o > 0; o >>= 1){ if (tid < o) red[tid] += red[tid+o]; __syncthreads(); }
    float csum = red[0];  __syncthreads();
    smacc += ssum / fmaxf(csum, 1.0f);
  }
  if (tid == 0){
    out[(size_t)78*SPIX]     = smacc / (float)NR;
    out[(size_t)78*SPIX + 1] = (float)(*touch) / (float)(NR*SPIX);
  }
}

// ---------------- launch ----------------
extern "C" void kernel_launch(void* const* d_in, const int* in_sizes, int n_in,
                              void* d_out, int out_size, void* d_ws, size_t ws_size,
                              hipStream_t stream)
{
  (void)in_sizes; (void)n_in; (void)out_size; (void)ws_size;
  const float* centers = (const float*)d_in[0];
  const float* scale   = (const float*)d_in[1];
  const float* feat    = (const float*)d_in[2];
  const float* opac    = (const float*)d_in[3];
  const float* bgp     = (const float*)d_in[4];
  const float* sem     = (const float*)d_in[5];
  const float* intr    = (const float*)d_in[6];
  const float* c2w     = (const float*)d_in[7];
  const float* fpw     = (const float*)d_in[8];
  float* out = (float*)d_out;

  float* W0 = (float*)d_ws;
  float*    AOS   = W0 + OFF_AOS;
  float*    MATS  = W0 + OFF_MATS;
  float*    U     = W0 + OFF_U;
  float*    Vv    = W0 + OFF_V;
  float*    SIG   = W0 + OFF_SIG;
  float*    WSIG  = W0 + OFF_WSIG;
  float*    WCNT  = W0 + OFF_WCNT;
  float*    KEYS  = W0 + OFF_KEYS;
  int*      SIDX  = (int*)(W0 + OFF_SIDX);
  unsigned* TOUCH = (unsigned*)(W0 + OFF_TOUCH);

  // passthrough sem_proj_2d -> output slot 7
  hipMemcpyAsync(out + (size_t)72*SPIX, sem, (size_t)NR*SPIX*sizeof(float),
                 hipMemcpyDeviceToDevice, stream);

  k_zero   <<<1, 32, 0, stream>>>(TOUCH);
  k_mats   <<<1, 32, 0, stream>>>(c2w, fpw, MATS);
  k_project<<<NR*24, 256, 0, stream>>>(centers, scale, opac, intr, MATS,
                                       U, Vv, SIG, WSIG, WCNT, KEYS);
  k_sort   <<<NR, 1024, 0, stream>>>(KEYS, SIDX);
  k_gather <<<(NR*NG)/256, 256, 0, stream>>>(SIDX, U, Vv, SIG, feat, opac, bgp, AOS);
  k_render <<<dim3(WW/16, HH/16, NR), dim3(16,16), 0, stream>>>(AOS, out, TOUCH);
  k_stats  <<<1, 256, 0, stream>>>(WSIG, WCNT, TOUCH, out);
}